// MyGCN_75385265979976
// MI455X (gfx1250) — compile-verified
//
#include <hip/hip_runtime.h>

typedef __attribute__((ext_vector_type(2))) float v2f;
typedef __attribute__((ext_vector_type(8))) float v8f;

#define N_NODES 50000
#define F_DIM   128
#define H_DIM   128
#define C_DIM   40

// ---------------------------------------------------------------- zero fill
__global__ void k_zero(float* __restrict__ p, long n) {
  long i = (long)blockIdx.x * blockDim.x + threadIdx.x;
  long stride = (long)gridDim.x * blockDim.x;
  for (; i < n; i += stride) p[i] = 0.0f;
}

// ---------------------------------------------------------------- degree count (target/col nodes)
__global__ void k_deg(const long long* __restrict__ col, float* __restrict__ deg, int E) {
  int e = blockIdx.x * blockDim.x + threadIdx.x;
  if (e < E) atomicAdd(&deg[col[e]], 1.0f);
}

// ---------------------------------------------------------------- dinv = rsqrt(deg + 1 self-loop)
__global__ void k_dinv(float* __restrict__ deg_dinv, int n) {
  int i = blockIdx.x * blockDim.x + threadIdx.x;
  if (i < n) {
    float d = deg_dinv[i] + 1.0f;   // self-loop => always > 0
    deg_dinv[i] = rsqrtf(d);
  }
}

// ---------------------------------------------------------------- GEMM1: xw = x @ W  (N x 128) via V_WMMA_F32_16X16X4_F32
// 8 waves / block, wave w owns the 16-column tile w; block owns a 16-row stripe.
__global__ void __launch_bounds__(256)
k_gemm1(const float* __restrict__ x, const float* __restrict__ W, float* __restrict__ xw) {
  const int lane = threadIdx.x & 31;
  const int wave = threadIdx.x >> 5;       // 0..7 -> column tile
  const int m    = lane & 15;
  const int g    = lane >> 4;              // 0 or 1
  const int row0 = blockIdx.x * 16;
  const int col0 = wave * 16;

  const float* xr = x + (long)(row0 + m) * F_DIM;   // A: row m of the tile
  const float* wc = W + col0 + m;                   // B: column m of the tile

  v8f acc = {};
#pragma unroll
  for (int k = 0; k < F_DIM; k += 4) {
    const int ka = k + 2 * g;
    v2f a; a.x = xr[ka];                  a.y = xr[ka + 1];
    v2f b; b.x = wc[(long)ka * H_DIM];    b.y = wc[(long)(ka + 1) * H_DIM];
    acc = __builtin_amdgcn_wmma_f32_16x16x4_f32(false, a, false, b, (short)0, acc, false, false);
  }

  float* o = xw + (long)row0 * H_DIM + col0 + m;
#pragma unroll
  for (int r = 0; r < 8; ++r)
    o[(long)(r + 8 * g) * H_DIM] = acc[r];          // acc[r] = D[r+8g][m]
}

// ---------------------------------------------------------------- edge scatter: agg[col] += xw[row] * dinv[row]*dinv[col]
// one wave per edge; each lane moves one float4 (4 global f32 atomics)
__global__ void __launch_bounds__(256)
k_scatter(const long long* __restrict__ row, const long long* __restrict__ col,
          const float* __restrict__ dinv, const float* __restrict__ xw,
          float* __restrict__ agg, int E) {
  const int lane = threadIdx.x & 31;
  const int e = (int)(((long)blockIdx.x * blockDim.x + threadIdx.x) >> 5);
  if (e >= E) return;
  const long long r = row[e];
  const long long c = col[e];
  const float nrm = dinv[r] * dinv[c];
  const float4 v = ((const float4*)(xw + (long)r * H_DIM))[lane];
  float* dst = agg + (long)c * H_DIM + lane * 4;
  atomicAdd(dst + 0, v.x * nrm);
  atomicAdd(dst + 1, v.y * nrm);
  atomicAdd(dst + 2, v.z * nrm);
  atomicAdd(dst + 3, v.w * nrm);
}

// ---------------------------------------------------------------- h = relu(agg + xw*dinv^2 (self-loop) + b); in place on d_out
__global__ void __launch_bounds__(256)
k_relu_bias(float* __restrict__ h_agg, const float* __restrict__ xw,
            const float* __restrict__ dinv, const float* __restrict__ bias, long total4) {
  long i = (long)blockIdx.x * blockDim.x + threadIdx.x;
  if (i >= total4) return;
  const long node = i >> 5;                 // 32 float4 per 128-wide row
  const int  q    = (int)(i & 31);
  const float di = dinv[node];
  const float self = di * di;
  float4 av = ((float4*)h_agg)[i];
  const float4 xv = ((const float4*)xw)[i];
  const float4 bv = ((const float4*)bias)[q];
  float4 r;
  r.x = fmaxf(av.x + xv.x * self + bv.x, 0.0f);
  r.y = fmaxf(av.y + xv.y * self + bv.y, 0.0f);
  r.z = fmaxf(av.z + xv.z * self + bv.z, 0.0f);
  r.w = fmaxf(av.w + xv.w * self + bv.w, 0.0f);
  ((float4*)h_agg)[i] = r;
}

// ---------------------------------------------------------------- GEMM2: out = h @ W_lin^T + b_lin  (N x 40, padded to 48)
// 3 waves / block = 3 column tiles of 16 (cols 40..47 masked to zero arithmetically: EXEC stays all-ones at WMMA)
__global__ void __launch_bounds__(96)
k_gemm2(const float* __restrict__ h, const float* __restrict__ Wl,
        const float* __restrict__ bl, float* __restrict__ out) {
  const int lane = threadIdx.x & 31;
  const int wave = threadIdx.x >> 5;        // 0..2
  const int m    = lane & 15;
  const int g    = lane >> 4;
  const int row0 = blockIdx.x * 16;
  const int c    = wave * 16 + m;           // output column 0..47
  const int cc   = (c < C_DIM) ? c : 0;
  const float cmask = (c < C_DIM) ? 1.0f : 0.0f;

  const float* hr = h + (long)(row0 + m) * H_DIM;   // A: row
  const float* wr = Wl + (long)cc * H_DIM;          // B[k][c] = W_lin[c][k]

  v8f acc = {};
#pragma unroll
  for (int k = 0; k < H_DIM; k += 4) {
    const int ka = k + 2 * g;
    v2f a; a.x = hr[ka];         a.y = hr[ka + 1];
    v2f b; b.x = wr[ka] * cmask; b.y = wr[ka + 1] * cmask;
    acc = __builtin_amdgcn_wmma_f32_16x16x4_f32(false, a, false, b, (short)0, acc, false, false);
  }

  if (c < C_DIM) {
    const float bb = bl[c];
#pragma unroll
    for (int r = 0; r < 8; ++r)
      out[(long)(row0 + r + 8 * g) * C_DIM + c] = acc[r] + bb;
  }
}

// ================================================================ host launcher
extern "C" void kernel_launch(void* const* d_in, const int* in_sizes, int n_in,
                              void* d_out, int out_size, void* d_ws, size_t ws_size,
                              hipStream_t stream) {
  const float*     x   = (const float*)d_in[0];
  const long long* ei  = (const long long*)d_in[1];   // int64 [2, E]
  const float*     Wg  = (const float*)d_in[2];
  const float*     bg  = (const float*)d_in[3];
  const float*     Wl  = (const float*)d_in[4];
  const float*     bl  = (const float*)d_in[5];

  const int E = in_sizes[1] / 2;
  const long long* row = ei;          // source
  const long long* col = ei + E;      // target

  // workspace: [dinv: N][xw: N*H]
  float* dinv = (float*)d_ws;
  float* xw   = dinv + N_NODES;

  // output: [h: N*H][out: N*C]; agg accumulates directly into the h slot
  float* h_out = (float*)d_out;
  float* out   = h_out + (long)N_NODES * H_DIM;

  const long nh = (long)N_NODES * H_DIM;

  // 1) zero dinv and the agg/h region (d_out is poisoned by the harness)
  k_zero<<<1024, 256, 0, stream>>>(dinv, (long)N_NODES);
  k_zero<<<4096, 256, 0, stream>>>(h_out, nh);

  // 2) degree of target nodes
  k_deg<<<(E + 255) / 256, 256, 0, stream>>>(col, dinv, E);

  // 3) dinv = rsqrt(deg + 1)
  k_dinv<<<(N_NODES + 255) / 256, 256, 0, stream>>>(dinv, N_NODES);

  // 4) xw = x @ W_gcn   (WMMA f32)
  k_gemm1<<<N_NODES / 16, 256, 0, stream>>>(x, Wg, xw);

  // 5) scatter-add messages over edges (one wave per edge)
  {
    const long waves = (long)E;
    const long blocks = (waves * 32 + 255) / 256;
    k_scatter<<<(int)blocks, 256, 0, stream>>>(row, col, dinv, xw, h_out, E);
  }

  // 6) h = relu(agg + self-loop + bias), in place in d_out
  {
    const long total4 = (long)N_NODES * (H_DIM / 4);
    k_relu_bias<<<(int)((total4 + 255) / 256), 256, 0, stream>>>(h_out, xw, dinv, bg, total4);
  }

  // 7) out = h @ W_lin^T + b_lin   (WMMA f32)
  k_gemm2<<<N_NODES / 16, 96, 0, stream>>>(h_out, Wl, bl, out);
}